// RegionProposalNetwork_8177617731632
// MI455X (gfx1250) — compile-verified
//
#include <hip/hip_runtime.h>
#include <hip/hip_bf16.h>

// ---------------------------------------------------------------------------
// RPN forward for MI455X (gfx1250, wave32).
// Pipeline: score-bits -> per-level exact top-1000 (radix select on 64-bit
// keys) -> per-batch decode + bitonic sort (LDS) -> pairwise IoU suppression
// bitmask (16x16 tiles, one wave each, union base via V_WMMA_F32_16X16X4_F32,
// bits packed with wave32 ballot) -> sequential greedy pass over bitmask ->
// compacting scatter of <=1000 survivors.
// ---------------------------------------------------------------------------

#define B_ 16
#define NTOT 159375            // 120000 + 30000 + 7500 + 1875
#define NC 4000                // 4 levels * 1000 candidates
#define MASKW 128              // 4096 bits per mask row (u32 words)
#define BBOX_CLIP 4.1351666f   // log(1000/16)
#define IMG 800.0f
#define OFFMULT 801.0f
#define NMS_T 0.7f

typedef __attribute__((ext_vector_type(2))) float v2f;
typedef __attribute__((ext_vector_type(8))) float v8f;

__device__ __forceinline__ unsigned f2s(float f) {
  unsigned u = __float_as_uint(f);
  return u ^ (((int)u >> 31) | 0x80000000u);
}
__device__ __forceinline__ float s2f(unsigned s) {
  unsigned b = (s & 0x80000000u) ? (s ^ 0x80000000u) : ~s;
  return __uint_as_float(b);
}
__device__ __forceinline__ void level_info(int n, int& l, int& idx, int& h, int& w) {
  if (n < 120000)      { l = 0; idx = n;          h = 200; w = 200; }
  else if (n < 150000) { l = 1; idx = n - 120000; h = 100; w = 100; }
  else if (n < 157500) { l = 2; idx = n - 150000; h = 50;  w = 50;  }
  else                 { l = 3; idx = n - 157500; h = 25;  w = 25;  }
}

// ---------------------------------------------------------------------------
// K1: objectness scores -> monotonic sortable u32 bits. obj[b, n] with
// n = (y*w + x)*A + a (matches _flat layout).
// ---------------------------------------------------------------------------
__global__ __launch_bounds__(256) void k1_scorebits(
    const float* __restrict__ c0, const float* __restrict__ c1,
    const float* __restrict__ c2, const float* __restrict__ c3,
    unsigned* __restrict__ objbits) {
  int gid = blockIdx.x * 256 + threadIdx.x;
  if (gid >= B_ * NTOT) return;
  int b = gid / NTOT, n = gid % NTOT;
  int l, idx, h, w;
  level_info(n, l, idx, h, w);
  const float* cl = (l == 0) ? c0 : (l == 1) ? c1 : (l == 2) ? c2 : c3;
  int a = idx % 3, hw = idx / 3, x = hw % w, y = hw / w;
  float v = cl[(((size_t)b * 3 + a) * h + y) * (size_t)w + x];
  objbits[gid] = f2s(v);
}

// ---------------------------------------------------------------------------
// K2: exact per-(batch, level) top-1000. 64-bit key = scorebits<<32 | ~idx
// (unique keys -> deterministic, reproduces lax.top_k tie order). 64-step
// bitwise radix select via repeated counting passes — the level's score
// array (<=480 KB) is L2-resident on MI455X's 192 MB L2, so passes are cheap.
// ---------------------------------------------------------------------------
__global__ __launch_bounds__(256) void k2_topk(
    const unsigned* __restrict__ objbits, unsigned* __restrict__ sel) {
  __shared__ unsigned s_cnt;
  int b = blockIdx.x >> 2, l = blockIdx.x & 3, tid = threadIdx.x;
  const int OFF[4] = {0, 120000, 150000, 157500};
  const int NL[4]  = {120000, 30000, 7500, 1875};
  int off = OFF[l], nl = NL[l];
  const unsigned* ob = objbits + (size_t)b * NTOT + off;

  unsigned long long prefix = 0ull;
  for (int bit = 63; bit >= 0; --bit) {
    unsigned long long cand = prefix | (1ull << bit);
    if (tid == 0) s_cnt = 0;
    __syncthreads();
    unsigned local = 0;
    for (int e = tid; e < nl; e += 256) {
      unsigned long long key =
          ((unsigned long long)ob[e] << 32) |
          (unsigned long long)(0xFFFFFFFFu - (unsigned)e);
      if (key >= cand) ++local;
    }
    atomicAdd(&s_cnt, local);
    __syncthreads();
    unsigned total = s_cnt;
    if (total >= 1000u) prefix = cand;
    __syncthreads();
  }
  // gather exactly the 1000 keys >= prefix (keys unique)
  if (tid == 0) s_cnt = 0;
  __syncthreads();
  for (int e = tid; e < nl; e += 256) {
    unsigned long long key =
        ((unsigned long long)ob[e] << 32) |
        (unsigned long long)(0xFFFFFFFFu - (unsigned)e);
    if (key >= prefix) {
      unsigned p = atomicAdd(&s_cnt, 1u);
      if (p < 1000u) sel[(size_t)b * NC + l * 1000 + p] = (unsigned)(off + e);
    }
  }
}

// ---------------------------------------------------------------------------
// K3: per-batch decode (box transform + clip + valid) and stable descending
// sort of 4000 candidates by (score||~n) 64-bit key — bitonic on 4096 slots
// in 48 KB LDS.
// ---------------------------------------------------------------------------
__global__ __launch_bounds__(256) void k3_decode_sort(
    const unsigned* __restrict__ objbits, const unsigned* __restrict__ sel,
    const float* __restrict__ r0, const float* __restrict__ r1,
    const float* __restrict__ r2, const float* __restrict__ r3,
    const float* __restrict__ anchors,
    float* __restrict__ cbox, float* __restrict__ coff,
    float* __restrict__ cscore, unsigned* __restrict__ cvalid,
    float* __restrict__ carea,
    float* __restrict__ sbox, float* __restrict__ soff,
    float* __restrict__ sscore, unsigned* __restrict__ svalid,
    float* __restrict__ sarea) {
  __shared__ unsigned long long skey[4096];
  __shared__ unsigned spay[4096];
  int b = blockIdx.x, tid = threadIdx.x;

  for (int c = tid; c < 4096; c += 256) {
    if (c < NC) {
      int n = (int)sel[(size_t)b * NC + c];
      float obj = s2f(objbits[(size_t)b * NTOT + n]);
      float score = 1.0f / (1.0f + expf(-obj));
      int l, idx, h, w;
      level_info(n, l, idx, h, w);
      const float* rg = (l == 0) ? r0 : (l == 1) ? r1 : (l == 2) ? r2 : r3;
      int a = idx % 3, hw = idx / 3, x = hw % w, y = hw / w;
      size_t hwsz = (size_t)h * w;
      size_t base = ((size_t)b * 12 + a * 4) * hwsz + (size_t)y * w + x;
      float d0 = rg[base], d1 = rg[base + hwsz];
      float d2 = fminf(rg[base + 2 * hwsz], BBOX_CLIP);
      float d3 = fminf(rg[base + 3 * hwsz], BBOX_CLIP);
      float a0 = anchors[(size_t)n * 4 + 0], a1 = anchors[(size_t)n * 4 + 1];
      float a2 = anchors[(size_t)n * 4 + 2], a3 = anchors[(size_t)n * 4 + 3];
      float wa = a2 - a0, ha = a3 - a1;
      float cxa = a0 + 0.5f * wa, cya = a1 + 0.5f * ha;
      float pcx = d0 * wa + cxa, pcy = d1 * ha + cya;
      float pw = expf(d2) * wa, ph = expf(d3) * ha;
      float x1 = fminf(fmaxf(pcx - 0.5f * pw, 0.f), IMG);
      float y1 = fminf(fmaxf(pcy - 0.5f * ph, 0.f), IMG);
      float x2 = fminf(fmaxf(pcx + 0.5f * pw, 0.f), IMG);
      float y2 = fminf(fmaxf(pcy + 0.5f * ph, 0.f), IMG);
      unsigned valid =
          ((x2 - x1) >= 1.f) && ((y2 - y1) >= 1.f) && (score >= 0.f);
      float lo = OFFMULT * (float)l;
      size_t ci = (size_t)b * NC + c;
      cbox[ci * 4 + 0] = x1; cbox[ci * 4 + 1] = y1;
      cbox[ci * 4 + 2] = x2; cbox[ci * 4 + 3] = y2;
      coff[ci * 4 + 0] = x1 + lo; coff[ci * 4 + 1] = y1 + lo;
      coff[ci * 4 + 2] = x2 + lo; coff[ci * 4 + 3] = y2 + lo;
      cscore[ci] = score;
      cvalid[ci] = valid;
      carea[ci] = (x2 - x1) * (y2 - y1);
      float ks = valid ? score : -1.0f;
      skey[c] = ((unsigned long long)f2s(ks) << 32) |
                (unsigned long long)(0xFFFFFFFFu - (unsigned)n);
      spay[c] = (unsigned)c;
    } else {
      skey[c] = 0ull;
      spay[c] = 0u;
    }
  }
  __syncthreads();

  for (unsigned k = 2; k <= 4096; k <<= 1) {
    for (unsigned j = k >> 1; j > 0; j >>= 1) {
      for (unsigned i = tid; i < 4096; i += 256) {
        unsigned p = i ^ j;
        if (p > i) {
          bool desc = ((i & k) == 0);
          unsigned long long ka = skey[i], kb = skey[p];
          if (desc ? (ka < kb) : (ka > kb)) {
            skey[i] = kb; skey[p] = ka;
            unsigned t = spay[i]; spay[i] = spay[p]; spay[p] = t;
          }
        }
      }
      __syncthreads();
    }
  }

  for (int s = tid; s < NC; s += 256) {
    size_t src = (size_t)b * NC + spay[s];
    size_t dst = (size_t)b * NC + s;
    sbox[dst * 4 + 0] = cbox[src * 4 + 0];
    sbox[dst * 4 + 1] = cbox[src * 4 + 1];
    sbox[dst * 4 + 2] = cbox[src * 4 + 2];
    sbox[dst * 4 + 3] = cbox[src * 4 + 3];
    soff[dst * 4 + 0] = coff[src * 4 + 0];
    soff[dst * 4 + 1] = coff[src * 4 + 1];
    soff[dst * 4 + 2] = coff[src * 4 + 2];
    soff[dst * 4 + 3] = coff[src * 4 + 3];
    sscore[dst] = cscore[src];
    svalid[dst] = cvalid[src];
    sarea[dst] = carea[src];
  }
}

// ---------------------------------------------------------------------------
// K4: pairwise suppression bitmask. One wave per 16x16 tile of the
// upper-triangular 4000x4000 candidate grid. The 256 (i,j) pairs map onto
// the WMMA 16x16 f32 C/D layout (lane l, vgpr v <-> M = v + 8*(l>=16),
// N = l%16). Union base areas_i + areas_j is computed on the matrix unit via
// V_WMMA_F32_16X16X4_F32 with A = [areas_i | 1 | 0 | 0], B = [1 ; areas_j^T].
// The threshold test inter/denom > T (denom > 0 always) is folded to
// inter > T*denom — no divide in the hot loop. Strictly-upper tiles skip the
// per-pair j>i compare entirely. Row bits packed with wave32 ballot.
// ---------------------------------------------------------------------------
__global__ __launch_bounds__(32) void k4_mask(
    const float* __restrict__ soff, const float* __restrict__ sarea,
    unsigned* __restrict__ mask) {
  int ti = blockIdx.x, tj = blockIdx.y, b = blockIdx.z;
  if (tj < ti) return;
  int lane = threadIdx.x;
  __shared__ float bi[16][4];
  __shared__ float bj[16][4];
  __shared__ float ai[16];
  __shared__ float aj[16];
  int gi0 = ti * 16, gj0 = tj * 16;

  if (lane < 16) {
    size_t g = (size_t)b * NC + gi0 + lane;
    bi[lane][0] = soff[g * 4 + 0]; bi[lane][1] = soff[g * 4 + 1];
    bi[lane][2] = soff[g * 4 + 2]; bi[lane][3] = soff[g * 4 + 3];
    ai[lane] = sarea[g];
  } else {
    int r = lane - 16;
    size_t g = (size_t)b * NC + gj0 + r;
    bj[r][0] = soff[g * 4 + 0]; bj[r][1] = soff[g * 4 + 1];
    bj[r][2] = soff[g * 4 + 2]; bj[r][3] = soff[g * 4 + 3];
    aj[r] = sarea[g];
  }
  __syncthreads();

  // A (16x4): lanes 0-15 hold M=lane, K=0..1; lanes 16-31 hold K=2..3 (zero).
  // B (4x16): lanes 0-15 hold N=lane rows K=0..1; lanes 16-31 rows K=2..3 (zero).
  v2f A = {0.f, 0.f};
  v2f Bm = {0.f, 0.f};
  if (lane < 16) {
    A.x = ai[lane]; A.y = 1.0f;       // A[m][0]=areas_i[m], A[m][1]=1
    Bm.x = 1.0f;    Bm.y = aj[lane];  // B[0][n]=1,          B[1][n]=areas_j[n]
  }
  v8f C = {};
  v8f D = __builtin_amdgcn_wmma_f32_16x16x4_f32(
      false, A, false, Bm, (short)0, C, false, false);
  // D[m][n] = areas_i[m] + areas_j[n]

  const bool diag = (tj == ti);
  int n = lane & 15;
  int mhalf = (lane >> 4) << 3;  // +8 for upper half-wave
  unsigned sh = 16u * (unsigned)(tj & 1);
  unsigned word = (unsigned)(tj >> 1);
  for (int v = 0; v < 8; ++v) {
    int m = v + mhalf;
    float ix1 = fmaxf(bi[m][0], bj[n][0]);
    float iy1 = fmaxf(bi[m][1], bj[n][1]);
    float ix2 = fminf(bi[m][2], bj[n][2]);
    float iy2 = fminf(bi[m][3], bj[n][3]);
    float inter = fmaxf(ix2 - ix1, 0.f) * fmaxf(iy2 - iy1, 0.f);
    float uni = D[v] - inter;
    float denom = (uni > 0.f) ? uni : 1.0f;
    bool pred = (inter > NMS_T * denom);      // == (inter/denom > NMS_T)
    if (diag) pred = pred && (n > m);          // only j > i on the diagonal
    unsigned bal = __builtin_amdgcn_ballot_w32(pred);
    if (lane == 0) {
      unsigned rlo = bal & 0xFFFFu;       // row gi0+v
      unsigned rhi = bal >> 16;           // row gi0+v+8
      if (rlo)
        atomicOr(&mask[((size_t)b * NC + gi0 + v) * MASKW + word], rlo << sh);
      if (rhi)
        atomicOr(&mask[((size_t)b * NC + gi0 + v + 8) * MASKW + word], rhi << sh);
    }
  }
}

// ---------------------------------------------------------------------------
// K5: sequential greedy pass over the precomputed mask (exact reference
// semantics: suppression applied only when candidate i survives), then
// rank/compact the <=1000 kept boxes into d_out. 128 threads: one lane owns
// one 32-bit suppression word. Next mask row is prefetched
// (global_prefetch_b8) while the current OR completes.
// ---------------------------------------------------------------------------
__global__ __launch_bounds__(128) void k5_greedy(
    const unsigned* __restrict__ svalid, const unsigned* __restrict__ mask,
    const float* __restrict__ sbox, const float* __restrict__ sscore,
    float* __restrict__ outB, float* __restrict__ outS) {
  int b = blockIdx.x, tid = threadIdx.x;
  __shared__ unsigned suppw[MASKW];
  __shared__ unsigned keepw[MASKW];
  __shared__ unsigned short srank[NC];
  __shared__ int curKeep;
  suppw[tid] = 0u;
  keepw[tid] = 0u;
  __syncthreads();

  for (int i = 0; i < NC; ++i) {
    if (tid == 0) {
      unsigned w = (unsigned)i >> 5, bm = 1u << (i & 31);
      int k = (svalid[(size_t)b * NC + i] != 0u) && ((suppw[w] & bm) == 0u);
      curKeep = k;
      if (k) keepw[w] |= bm;
    }
    __syncthreads();
    if (curKeep) suppw[tid] |= mask[((size_t)b * NC + i) * MASKW + tid];
    if (i + 1 < NC)
      __builtin_prefetch(&mask[((size_t)b * NC + i + 1) * MASKW + tid], 0, 0);
    __syncthreads();
  }

  if (tid == 0) {
    int r = 0;
    for (int i = 0; i < NC; ++i) {
      if (keepw[i >> 5] & (1u << (i & 31))) {
        srank[i] = (r < 1000) ? (unsigned short)r : (unsigned short)0xFFFF;
        ++r;
      } else {
        srank[i] = 0xFFFF;
      }
    }
  }
  __syncthreads();

  for (int i = tid; i < NC; i += 128) {
    unsigned short r = srank[i];
    if (r != 0xFFFF) {
      size_t si = (size_t)b * NC + i;
      size_t di = (size_t)b * 1000 + r;
      outB[di * 4 + 0] = sbox[si * 4 + 0];
      outB[di * 4 + 1] = sbox[si * 4 + 1];
      outB[di * 4 + 2] = sbox[si * 4 + 2];
      outB[di * 4 + 3] = sbox[si * 4 + 3];
      outS[di] = sscore[si];
    }
  }
}

// ---------------------------------------------------------------------------
extern "C" void kernel_launch(void* const* d_in, const int* in_sizes, int n_in,
                              void* d_out, int out_size, void* d_ws,
                              size_t ws_size, hipStream_t stream) {
  (void)in_sizes; (void)n_in; (void)ws_size;
  const float* c0 = (const float*)d_in[0];
  const float* c1 = (const float*)d_in[1];
  const float* c2 = (const float*)d_in[2];
  const float* c3 = (const float*)d_in[3];
  const float* r0 = (const float*)d_in[4];
  const float* r1 = (const float*)d_in[5];
  const float* r2 = (const float*)d_in[6];
  const float* r3 = (const float*)d_in[7];
  const float* anchors = (const float*)d_in[8];

  char* w = (char*)d_ws;
  auto take = [&](size_t bytes) -> void* {
    void* p = (void*)w;
    w += (bytes + 255) & ~(size_t)255;
    return p;
  };
  unsigned* objbits = (unsigned*)take((size_t)B_ * NTOT * 4);        // 10.2 MB
  unsigned* sel     = (unsigned*)take((size_t)B_ * NC * 4);
  float*    cbox    = (float*)take((size_t)B_ * NC * 16);
  float*    coff    = (float*)take((size_t)B_ * NC * 16);
  float*    cscore  = (float*)take((size_t)B_ * NC * 4);
  unsigned* cvalid  = (unsigned*)take((size_t)B_ * NC * 4);
  float*    carea   = (float*)take((size_t)B_ * NC * 4);
  float*    sbox    = (float*)take((size_t)B_ * NC * 16);
  float*    soff    = (float*)take((size_t)B_ * NC * 16);
  float*    sscore  = (float*)take((size_t)B_ * NC * 4);
  unsigned* svalid  = (unsigned*)take((size_t)B_ * NC * 4);
  float*    sarea   = (float*)take((size_t)B_ * NC * 4);
  size_t maskBytes  = (size_t)B_ * NC * MASKW * 4;                   // 32.8 MB
  unsigned* mask    = (unsigned*)take(maskBytes);

  float* outB = (float*)d_out;                // (B, 1000, 4)
  float* outS = outB + (size_t)B_ * 1000 * 4; // (B, 1000)

  hipMemsetAsync(d_out, 0, (size_t)out_size * sizeof(float), stream);
  hipMemsetAsync(mask, 0, maskBytes, stream);

  int tot = B_ * NTOT;
  k1_scorebits<<<(tot + 255) / 256, 256, 0, stream>>>(c0, c1, c2, c3, objbits);
  k2_topk<<<B_ * 4, 256, 0, stream>>>(objbits, sel);
  k3_decode_sort<<<B_, 256, 0, stream>>>(objbits, sel, r0, r1, r2, r3, anchors,
                                         cbox, coff, cscore, cvalid, carea,
                                         sbox, soff, sscore, svalid, sarea);
  dim3 g4(NC / 16, NC / 16, B_);
  k4_mask<<<g4, 32, 0, stream>>>(soff, sarea, mask);
  k5_greedy<<<B_, 128, 0, stream>>>(svalid, mask, sbox, sscore, outB, outS);
}